// ImprovedGCN_9440338117505
// MI455X (gfx1250) — compile-verified
//
#include <hip/hip_runtime.h>
#include <hip/hip_bf16.h>
#include <stdint.h>

typedef __attribute__((ext_vector_type(2))) float v2f;
typedef __attribute__((ext_vector_type(8))) float v8f;

#define D      128
#define LAYERS 4
#define NGRAPH 64
#define HID2   64
#define EPSV   1e-5f

// ---------------- utility ----------------
__global__ __launch_bounds__(256) void zero_f32(float* __restrict__ p, long n) {
  long i = (long)blockIdx.x * blockDim.x + threadIdx.x;
  long stride = (long)gridDim.x * blockDim.x;
  for (; i < n; i += stride) p[i] = 0.0f;
}

// ---------------- degree / norm ----------------
__global__ __launch_bounds__(256) void deg_kernel(const int* __restrict__ dst,
                                                  float* __restrict__ deg, int E) {
  int i = blockIdx.x * blockDim.x + threadIdx.x;
  if (i < E) atomicAdd(&deg[dst[i]], 1.0f);
}

__global__ __launch_bounds__(256) void dinv_kernel(float* __restrict__ deg, int N) {
  int i = blockIdx.x * blockDim.x + threadIdx.x;
  if (i < N) deg[i] = rsqrtf(deg[i] + 1.0f);   // +1 = self loop; deg >= 1 always
}

// ---------------- transpose all layer weights: Wt[l][j][i] = W[l][i][j] ----------------
__global__ __launch_bounds__(256) void transpose_w_kernel(const float* __restrict__ W,
                                                          float* __restrict__ Wt) {
  int idx = blockIdx.x * blockDim.x + threadIdx.x;    // over LAYERS*D*D
  if (idx >= LAYERS * D * D) return;
  int l = idx / (D * D);
  int r = idx - l * D * D;
  int i = r / D, j = r % D;
  Wt[(long)l * D * D + (long)j * D + i] = W[(long)l * D * D + (long)i * D + j];
}

// ---------------- GEMM: H = X @ W + bias  (fp32 WMMA 16x16x4) ----------------
// Block = 256 threads = 8 waves; wave w owns output columns [16w, 16w+16).
// B tile (128x16 per wave) is preloaded into 64 VGPRs from the transposed
// weights; blocks are persistent over row tiles so B loads amortize away and
// the steady-state inner loop is ds_load_2addr_b64 + v_wmma only.
__global__ __launch_bounds__(256) void gemm_bias_wmma(
    const float* __restrict__ X, const float* __restrict__ Wt,
    const float* __restrict__ bias, float* __restrict__ H,
    int n, int numTiles) {
  __shared__ float As[16][D + 4];          // +4 pad: rows hit distinct LDS banks
  const int tid  = threadIdx.x;
  const int lane = tid & 31;
  const int wave = tid >> 5;
  const int col_base = wave * 16;
  const int m    = lane & 15;              // row (A) / col (B) within tile
  const int half = lane >> 4;              // K sub-pair select

  // Preload B: lane (m, half) needs column (col_base+m) of W at
  // k = 4s + 2*half, 4s + 2*half + 1 for s = 0..31  -> contiguous float2 loads.
  v2f breg[32];
  {
    const float* wcol = Wt + (long)(col_base + m) * D + half * 2;
    #pragma unroll
    for (int s = 0; s < 32; ++s) {
      float2 bb = *reinterpret_cast<const float2*>(wcol + 4 * s);
      breg[s].x = bb.x;
      breg[s].y = bb.y;
    }
  }
  const float bv = bias[col_base + m];

  for (int t = blockIdx.x; t < numTiles; t += gridDim.x) {
    const int row_base = t * 16;

    // Cooperative coalesced load of the 16x128 A tile (512 float4, 2/thread).
    #pragma unroll
    for (int i = 0; i < 2; ++i) {
      int f  = tid + i * 256;              // float4 index
      int r  = f >> 5;                     // 32 float4 per row
      int c4 = f & 31;
      float4 v = make_float4(0.f, 0.f, 0.f, 0.f);
      int grow = row_base + r;
      if (grow < n) v = *reinterpret_cast<const float4*>(X + (long)grow * D + c4 * 4);
      As[r][c4 * 4 + 0] = v.x; As[r][c4 * 4 + 1] = v.y;
      As[r][c4 * 4 + 2] = v.z; As[r][c4 * 4 + 3] = v.w;
    }
    __syncthreads();

    v8f c = {};
    #pragma unroll
    for (int s = 0; s < 32; ++s) {
      const int kk = 4 * s + half * 2;
      v2f a;
      a.x = As[m][kk];
      a.y = As[m][kk + 1];
      c = __builtin_amdgcn_wmma_f32_16x16x4_f32(false, a, false, breg[s],
                                                (short)0, c, false, false);
    }

    #pragma unroll
    for (int r = 0; r < 8; ++r) {
      int grow = row_base + r + half * 8;  // C/D layout: VGPR r -> rows r, r+8
      if (grow < n) H[(long)grow * D + col_base + m] = c[r] + bv;
    }
    __syncthreads();                       // As reused next iteration
  }
}

// ---------------- scatter-add: agg[dst] += norm * h[src] (one wave per edge) ----------------
__global__ __launch_bounds__(256) void scatter_kernel(
    const int* __restrict__ src, const int* __restrict__ dst,
    const float* __restrict__ dinv, const float* __restrict__ H,
    float* __restrict__ AGG, int E, int N) {
  int e    = blockIdx.x * 8 + (threadIdx.x >> 5);
  int lane = threadIdx.x & 31;
  int T = E + N;
  if (e >= T) return;
  int s, d; float w;
  if (e < E) {
    s = src[e]; d = dst[e];
    w = dinv[s] * dinv[d];
  } else {
    s = e - E; d = s;
    float dv = dinv[s];
    w = dv * dv;                           // self-loop norm
  }
  float4 v = *reinterpret_cast<const float4*>(H + (long)s * D + lane * 4);
  float* out = AGG + (long)d * D + lane * 4;
  atomicAdd(out + 0, v.x * w);
  atomicAdd(out + 1, v.y * w);
  atomicAdd(out + 2, v.z * w);
  atomicAdd(out + 3, v.w * w);
}

// ---------------- BN stats: per-column sum / sumsq ----------------
__global__ __launch_bounds__(256) void stats_kernel(const float* __restrict__ AGG,
                                                    float* __restrict__ stats, int N) {
  long total  = (long)N * D;
  long i      = (long)blockIdx.x * blockDim.x + threadIdx.x;
  long stride = (long)gridDim.x * blockDim.x;   // multiple of D (blockDim=256)
  int col = (int)(i & (D - 1));
  float s = 0.f, s2 = 0.f;
  for (; i < total; i += stride) {
    float v = AGG[i];
    s += v; s2 += v * v;
  }
  atomicAdd(&stats[col],     s);
  atomicAdd(&stats[D + col], s2);
}

__global__ void bn_coef_kernel(const float* __restrict__ stats,
                               const float* __restrict__ gamma,
                               const float* __restrict__ beta,
                               float* __restrict__ coef, int N) {
  int c = threadIdx.x;                     // launched with D threads
  float invN = 1.0f / (float)N;
  float mean = stats[c] * invN;
  float var  = stats[D + c] * invN - mean * mean;
  float inv  = rsqrtf(var + EPSV);
  float sc   = gamma[c] * inv;
  coef[c]     = sc;                        // scale
  coef[D + c] = beta[c] - mean * sc;       // shift
}

// ---------------- BN apply + ReLU + residual ----------------
__global__ __launch_bounds__(256) void bn_apply_kernel(
    const float* __restrict__ AGG, const float* __restrict__ XIN,
    const float* __restrict__ coef, float* __restrict__ XOUT, int N) {
  long total  = (long)N * D;
  long i      = (long)blockIdx.x * blockDim.x + threadIdx.x;
  long stride = (long)gridDim.x * blockDim.x;
  for (; i < total; i += stride) {
    int col = (int)(i & (D - 1));
    float v = fmaf(AGG[i], coef[col], coef[D + col]);
    v = fmaxf(v, 0.0f);
    XOUT[i] = v + XIN[i];
  }
}

// ---------------- global mean pool (wave per node) ----------------
__global__ __launch_bounds__(256) void pool_kernel(
    const float* __restrict__ X, const int* __restrict__ batch,
    float* __restrict__ pool, float* __restrict__ cnt, int N) {
  int nidx = blockIdx.x * 8 + (threadIdx.x >> 5);
  int lane = threadIdx.x & 31;
  if (nidx >= N) return;
  int g = batch[nidx];
  float4 v = *reinterpret_cast<const float4*>(X + (long)nidx * D + lane * 4);
  float* out = pool + (long)g * D + lane * 4;
  atomicAdd(out + 0, v.x);
  atomicAdd(out + 1, v.y);
  atomicAdd(out + 2, v.z);
  atomicAdd(out + 3, v.w);
  if (lane == 0) atomicAdd(&cnt[g], 1.0f);
}

// ---------------- head MLP ----------------
__global__ __launch_bounds__(256) void head1_kernel(
    const float* __restrict__ pool, const float* __restrict__ cnt,
    const float* __restrict__ W1, const float* __restrict__ b1,
    float* __restrict__ h1) {
  int idx = blockIdx.x * blockDim.x + threadIdx.x;   // 64*64
  if (idx >= NGRAPH * HID2) return;
  int g = idx / HID2, j = idx % HID2;
  float inv = 1.0f / fmaxf(cnt[g], 1.0f);
  float acc = b1[j];
  #pragma unroll 8
  for (int c = 0; c < D; ++c)
    acc = fmaf(pool[g * D + c] * inv, W1[c * HID2 + j], acc);
  h1[idx] = fmaxf(acc, 0.0f);
}

__global__ void head2_kernel(const float* __restrict__ h1,
                             const float* __restrict__ W2,
                             const float* __restrict__ b2,
                             float* __restrict__ out) {
  int idx = blockIdx.x * blockDim.x + threadIdx.x;   // 64*2
  if (idx >= NGRAPH * 2) return;
  int g = idx / 2, k = idx % 2;
  float acc = b2[k];
  #pragma unroll 8
  for (int j = 0; j < HID2; ++j)
    acc = fmaf(h1[g * HID2 + j], W2[j * 2 + k], acc);
  out[idx] = acc;
}

// ---------------- launch ----------------
extern "C" void kernel_launch(void* const* d_in, const int* in_sizes, int n_in,
                              void* d_out, int out_size, void* d_ws, size_t ws_size,
                              hipStream_t stream) {
  const float* x       = (const float*)d_in[0];
  const int*   edge    = (const int*)d_in[1];
  const int*   batch   = (const int*)d_in[2];
  const float* conv_w  = (const float*)d_in[3];
  const float* conv_b  = (const float*)d_in[4];
  const float* gamma   = (const float*)d_in[5];
  const float* beta    = (const float*)d_in[6];
  const float* head_w1 = (const float*)d_in[7];
  const float* head_b1 = (const float*)d_in[8];
  const float* head_w2 = (const float*)d_in[9];
  const float* head_b2 = (const float*)d_in[10];

  const int N = in_sizes[0] / D;
  const int E = in_sizes[1] / 2;
  const int* srcI = edge;        // edge_index[0]
  const int* dstI = edge + E;    // edge_index[1]
  const long nd = (long)N * D;

  float* ws     = (float*)d_ws;
  float* xbuf   = ws;  ws += nd;
  float* hbuf   = ws;  ws += nd;
  float* aggbuf = ws;  ws += nd;
  float* wt     = ws;  ws += (long)LAYERS * D * D;   // transposed conv weights
  float* dinv   = ws;  ws += N;  ws += (4 - (N & 3)) & 3;
  float* stats  = ws;  ws += 2 * D;
  float* coef   = ws;  ws += 2 * D;
  float* pool   = ws;  ws += NGRAPH * D;
  float* cnt    = ws;  ws += NGRAPH;      // contiguous after pool
  float* h1     = ws;  ws += NGRAPH * HID2;

  dim3 b256(256);

  // transposed weights (once per call)
  transpose_w_kernel<<<(LAYERS * D * D + 255) / 256, b256, 0, stream>>>(conv_w, wt);

  // GCN degree norm with self-loops
  zero_f32<<<512, b256, 0, stream>>>(dinv, N);
  deg_kernel<<<(E + 255) / 256, b256, 0, stream>>>(dstI, dinv, E);
  dinv_kernel<<<(N + 255) / 256, b256, 0, stream>>>(dinv, N);

  const int rowTiles = (N + 15) / 16;
  const int gemmGrid = rowTiles < 2048 ? rowTiles : 2048;   // persistent blocks
  const int T = E + N;
  for (int l = 0; l < LAYERS; ++l) {
    const float* xin = (l == 0) ? x : xbuf;
    gemm_bias_wmma<<<gemmGrid, b256, 0, stream>>>(
        xin, wt + (long)l * D * D, conv_b + (long)l * D, hbuf, N, rowTiles);
    zero_f32<<<2048, b256, 0, stream>>>(aggbuf, nd);
    scatter_kernel<<<(T + 7) / 8, b256, 0, stream>>>(srcI, dstI, dinv, hbuf,
                                                     aggbuf, E, N);
    zero_f32<<<1, b256, 0, stream>>>(stats, 2 * D);
    stats_kernel<<<512, b256, 0, stream>>>(aggbuf, stats, N);
    bn_coef_kernel<<<1, D, 0, stream>>>(stats, gamma + (long)l * D,
                                        beta + (long)l * D, coef, N);
    bn_apply_kernel<<<2048, b256, 0, stream>>>(aggbuf, xin, coef, xbuf, N);
  }

  // pooling + head
  zero_f32<<<64, b256, 0, stream>>>(pool, NGRAPH * D + NGRAPH);  // pool + cnt
  pool_kernel<<<(N + 7) / 8, b256, 0, stream>>>(xbuf, batch, pool, cnt, N);
  head1_kernel<<<(NGRAPH * HID2 + 255) / 256, b256, 0, stream>>>(
      pool, cnt, head_w1, head_b1, h1);
  head2_kernel<<<1, 128, 0, stream>>>(h1, head_w2, head_b2, (float*)d_out);
}